// AI4DEM_90778428768641
// MI455X (gfx1250) — compile-verified
//
#include <hip/hip_runtime.h>
#include <stdint.h>

// Problem constants (reference: NY = NX = 4096, DX = DY = 1, VP = 3.1415/6)
#define NXD   4096
#define NYD   4096
#define WMASK 4095              // power-of-two wrap for jnp.roll periodicity
#define TX    32                // tile width  (one wave row)
#define TY    16                // tile height (2 rows per thread, blockDim.y=8)
#define HX    (TX + 2)          // 34 halo cols
#define HY    (TY + 2)          // 18 halo rows
#define LSTR  36                // padded LDS row stride (floats)
#define NFLD  7                 // xp, yp, vx, vy, Fx, Fy, mask

typedef __attribute__((address_space(1))) int         as1_int;
typedef __attribute__((address_space(3))) int         as3_int;
typedef __attribute__((address_space(3))) float       as3_float;

// ---- CDNA5 async global->LDS copy (ASYNCcnt path) -------------------------
__device__ __forceinline__ void async_g2l_b32(const float* gptr, float* lptr) {
#if __has_builtin(__builtin_amdgcn_global_load_async_to_lds_b32)
    __builtin_amdgcn_global_load_async_to_lds_b32(
        (as1_int*)(uintptr_t)gptr, (as3_int*)lptr, /*offset=*/0, /*cpol=*/0);
#else
    // GV mode: 64-bit vaddr pair, LDS byte-offset in the "vdst" slot.
    uint32_t lofs = (uint32_t)(uintptr_t)(as3_float*)lptr;
    asm volatile("global_load_async_to_lds_b32 %0, %1, off"
                 :: "v"(lofs), "v"(gptr) : "memory");
#endif
}

__device__ __forceinline__ void wait_async_zero() {
#if __has_builtin(__builtin_amdgcn_s_wait_asynccnt)
    __builtin_amdgcn_s_wait_asynccnt(0);
#else
    asm volatile("s_wait_asynccnt 0" ::: "memory");
#endif
}

// ---------------------------------------------------------------------------
__global__ __launch_bounds__(256)
void ai4dem_coupling_kernel(const float* __restrict__ xp,
                            const float* __restrict__ yp,
                            const float* __restrict__ vx,
                            const float* __restrict__ vy,
                            const float* __restrict__ fx,
                            const float* __restrict__ fy,
                            const float* __restrict__ mk,
                            float* __restrict__ out)
{
    __shared__ float lds[NFLD][HY][LSTR];   // ~18 KB of 320 KB/WGP

    const int tid = threadIdx.y * TX + threadIdx.x;   // 0..255
    const int bx  = blockIdx.x * TX;
    const int by  = blockIdx.y * TY;

    const float* bases[NFLD] = { xp, yp, vx, vy, fx, fy, mk };

    // ---- Stage halo (HYxHX) of all 7 fields into LDS via async loads ----
    // Per-lane wrapped global addressing handles jnp.roll periodicity exactly.
    const int NEL = HY * HX;                          // 612 elements / field
#pragma unroll
    for (int a = 0; a < NFLD; ++a) {
        const float* src = bases[a];
        for (int e = tid; e < NEL; e += 256) {
            const int lr = e / HX;
            const int lc = e - lr * HX;
            const int gy = (by + lr - 1) & WMASK;
            const int gx = (bx + lc - 1) & WMASK;
            async_g2l_b32(src + (size_t)gy * NXD + gx, &lds[a][lr][lc]);
        }
    }
    wait_async_zero();      // this wave's async transfers landed in LDS
    __syncthreads();        // all waves' transfers landed

    // ---- Compute: each thread does 2 output rows of the 32x16 tile ----
    const int tx = threadIdx.x;
#pragma unroll
    for (int k = 0; k < 2; ++k) {
        const int r  = threadIdx.y + k * 8;           // local row 0..15
        const int gx = bx + tx;
        const int gy = by + r;
        const float xc = (float)gx;
        const float yc = (float)gy;

        float s_a = 0.f, s_u = 0.f, s_v = 0.f, s_fx = 0.f, s_fy = 0.f;

#pragma unroll
        for (int dj = 0; dj < 3; ++dj) {
#pragma unroll
            for (int di = 0; di < 3; ++di) {
                const int hr = r + dj;                // halo row (r-1+dj)+1
                const int hc = tx + di;               // halo col
                const float xpn = lds[0][hr][hc];
                const float ypn = lds[1][hr][hc];
                const float mkn = lds[6][hr][hc];

                const float sgx = xpn - xc;
                const float Sx  = (sgx >= 0.f) ? 1.f : -1.f;
                const float xii = 2.f * sgx - Sx;     // DX = 1
                const float inx = (xii >= -1.f && xii <= 1.f) ? 1.f : 0.f;
                const float Nx  = 0.5f * (1.f - Sx * xii) * inx;

                const float sgy = ypn - yc;
                const float Sy  = (sgy >= 0.f) ? 1.f : -1.f;
                const float phi = 2.f * sgy - Sy;     // DY = 1
                const float iny = (phi >= -1.f && phi <= 1.f) ? 1.f : 0.f;
                const float Ny  = 0.5f * (1.f - Sy * phi) * iny;

                const float w = Nx * Ny * mkn * (float)(3.1415 / 6.0); // *inv_area(=1)

                s_a  += w;
                s_u  += w * lds[2][hr][hc];
                s_v  += w * lds[3][hr][hc];
                s_fx += w * lds[4][hr][hc];
                s_fy += w * lds[5][hr][hc];
            }
        }

        const size_t N   = (size_t)NXD * NYD;
        const size_t idx = (size_t)gy * NXD + gx;
        out[idx]         = s_a;
        out[N + idx]     = s_u;
        out[2 * N + idx] = s_v;
        out[3 * N + idx] = s_fx;
        out[4 * N + idx] = s_fy;
    }
}

// ---------------------------------------------------------------------------
extern "C" void kernel_launch(void* const* d_in, const int* in_sizes, int n_in,
                              void* d_out, int out_size, void* d_ws, size_t ws_size,
                              hipStream_t stream) {
    // setup_inputs order: xp, yp, x_grid, y_grid, vx, vy, Fx, Fy, mask
    const float* xp = (const float*)d_in[0];
    const float* yp = (const float*)d_in[1];
    // d_in[2] (x_grid) / d_in[3] (y_grid) are pure coordinates -> recomputed
    const float* vx = (const float*)d_in[4];
    const float* vy = (const float*)d_in[5];
    const float* fx = (const float*)d_in[6];
    const float* fy = (const float*)d_in[7];
    const float* mk = (const float*)d_in[8];
    float* out = (float*)d_out;

    dim3 block(TX, 8, 1);                       // 256 threads = 8 wave32
    dim3 grid(NXD / TX, NYD / TY, 1);           // 128 x 256 blocks
    hipLaunchKernelGGL(ai4dem_coupling_kernel, grid, block, 0, stream,
                       xp, yp, vx, vy, fx, fy, mk, out);
}